// CRF_51522427683249
// MI455X (gfx1250) — compile-verified
//
#include <hip/hip_runtime.h>
#include <cmath>

typedef __attribute__((ext_vector_type(2))) float v2f;
typedef __attribute__((ext_vector_type(8))) float v8f;

#define TC   16                 // channels (fixed by reference, matches WMMA 16x16)
#define TH   16                 // output tile height
#define TW   32                 // output tile width
#define HALO 5                  // 11-tap conv halo
#define RH   (TH + 2*HALO)      // 26
#define RW   (TW + 2*HALO)      // 42
#define NPIX (TH*TW)            // 512 output pixels per block

// Fused CRF mean-field iteration:
//   p = softmax(x_src)  -> LDS (tile + halo, zero-padded)
//   t = conv_H(p, k0)   -> LDS
//   s = w * conv_W(t, k1)
//   c = s x comp  (comp = -I16) via V_WMMA_F32_16X16X4_F32
//   x_dst = x0 - c
__global__ __launch_bounds__(256) void crf_iter_kernel(
    const float* __restrict__ xsrc, const float* __restrict__ x0,
    float* __restrict__ xdst,
    const float* __restrict__ spacings, const float* __restrict__ inv_theta,
    const float* __restrict__ weight, int H, int W)
{
    extern __shared__ float smem[];
    float* pbuf = smem;                   // [TC][RH][RW]  softmax probs (+halo)
    float* tbuf = smem + TC*RH*RW;        // [TC][TH][RW]  after conv along H
    float* sbuf = smem;                   // [NPIX][TC]    s (aliases pbuf)
    float* obuf = smem + TC*RH*RW;        // [NPIX][TC]    s x comp (aliases tbuf)

    const int  tid = threadIdx.x;
    const int  b   = blockIdx.z;
    const int  h0  = blockIdx.y * TH;
    const int  w0  = blockIdx.x * TW;
    const long HW  = (long)H * W;

    // Per-thread Gaussian taps (cheap: 22 expf per thread). Center tap zeroed.
    float k0[11], k1w[11];
    {
        const float sp0 = spacings[2*b+0], sp1 = spacings[2*b+1];
        const float it0 = inv_theta[0],    it1 = inv_theta[1];
        const float wgt = weight[0];
        #pragma unroll
        for (int t = 0; t < 11; ++t) {
            float d0 = sp0 * (float)(t-5) * it0;
            float d1 = sp1 * (float)(t-5) * it1;
            k0[t]  = (t == 5) ? 0.f : expf(-0.5f*d0*d0);
            k1w[t] = (t == 5) ? 0.f : wgt * expf(-0.5f*d1*d1);
        }
    }

    const float* xb = xsrc + (long)b * TC * HW;

    // ---- Phase 1: softmax over C into pbuf; zero pad outside image -------
    for (int e = tid; e < RH*RW; e += 256) {
        int r = e / RW, cc = e % RW;
        int gh = h0 + r - HALO, gw = w0 + cc - HALO;
        if (gh >= 0 && gh < H && gw >= 0 && gw < W) {
            const float* px = xb + (long)gh * W + gw;
            float v[TC]; float m = -3.402823466e38f;
            #pragma unroll
            for (int c = 0; c < TC; ++c) { v[c] = px[c*HW]; m = fmaxf(m, v[c]); }
            float ssum = 0.f;
            #pragma unroll
            for (int c = 0; c < TC; ++c) { v[c] = expf(v[c] - m); ssum += v[c]; }
            float inv = 1.f / ssum;
            #pragma unroll
            for (int c = 0; c < TC; ++c) pbuf[(c*RH + r)*RW + cc] = v[c]*inv;
        } else {
            #pragma unroll
            for (int c = 0; c < TC; ++c) pbuf[(c*RH + r)*RW + cc] = 0.f;
        }
    }
    __syncthreads();

    // ---- Phase 2: 11-tap conv along H ------------------------------------
    for (int e = tid; e < TC*TH*RW; e += 256) {
        int c = e / (TH*RW), rem = e % (TH*RW);
        int r = rem / RW,    cc  = rem % RW;
        float acc = 0.f;
        #pragma unroll
        for (int t = 0; t < 11; ++t)
            acc = fmaf(k0[t], pbuf[(c*RH + r + t)*RW + cc], acc);
        tbuf[(c*TH + r)*RW + cc] = acc;
    }
    __syncthreads();

    // ---- Phase 3: 11-tap conv along W (weight folded in) -> sbuf[pix][c] -
    for (int e = tid; e < NPIX*TC; e += 256) {
        int c = e / NPIX, pix = e % NPIX;
        int r = pix / TW, cw  = pix % TW;
        float acc = 0.f;
        #pragma unroll
        for (int t = 0; t < 11; ++t)
            acc = fmaf(k1w[t], tbuf[(c*TH + r)*RW + cw + t], acc);
        sbuf[pix*TC + c] = acc;
    }
    __syncthreads();

    // ---- Phase 4: c = s x comp (comp = -I) via V_WMMA_F32_16X16X4_F32 ----
    // A: M=16 pixels x K=16 channels, chunked K=4. B: comp chunk (4x16).
    // A layout (f32 16x4): lanes 0-15 row M=lane K={4q,4q+1}; lanes 16-31
    // same M with K={4q+2,4q+3}.  B rows striped across lanes, same half-split.
    {
        const int lane = tid & 31;
        const int wv   = tid >> 5;
        const int half = (lane < 16) ? 0 : 1;
        const int mrow = lane & 15;
        for (int T = wv; T < NPIX/16; T += 8) {
            const float* sp = sbuf + T*16*TC;
            v8f d = {0.f,0.f,0.f,0.f,0.f,0.f,0.f,0.f};
            #pragma unroll
            for (int q = 0; q < 4; ++q) {
                int kx = 4*q + 2*half;     // K index held in vgpr0 for this lane
                int ky = kx + 1;           // K index held in vgpr1
                v2f a, bm;
                a.x  = sp[mrow*TC + kx];
                a.y  = sp[mrow*TC + ky];
                bm.x = (kx == mrow) ? -1.f : 0.f;   // comp[k][n] = -(k==n)
                bm.y = (ky == mrow) ? -1.f : 0.f;
                d = __builtin_amdgcn_wmma_f32_16x16x4_f32(
                        false, a, false, bm, (short)0, d, false, false);
            }
            // D layout: vgpr j -> (M=j, N=lane) lanes 0-15; (M=j+8, N=lane-16)
            #pragma unroll
            for (int j = 0; j < 8; ++j) {
                int Mo = j + 8*half;
                obuf[(T*16 + Mo)*TC + mrow] = d[j];
            }
        }
    }
    __syncthreads();

    // ---- Phase 5: x_dst = x0 - c  (coalesced along W) --------------------
    const float* x0b = x0   + (long)b * TC * HW;
    float*       xdb = xdst + (long)b * TC * HW;
    for (int e = tid; e < NPIX*TC; e += 256) {
        int c = e / NPIX, pix = e % NPIX;
        int r = pix / TW, cw  = pix % TW;
        int gh = h0 + r,  gw  = w0 + cw;
        if (gh < H && gw < W) {
            long idx = (long)c*HW + (long)gh*W + gw;
            xdb[idx] = x0b[idx] - obuf[pix*TC + c];
        }
    }
}

__global__ __launch_bounds__(256) void crf_logsoftmax_kernel(
    const float* __restrict__ x, float* __restrict__ y, long npix, long HW)
{
    long pix = (long)blockIdx.x * 256 + threadIdx.x;
    if (pix >= npix) return;
    long bb = pix / HW, rem = pix % HW;
    const float* px = x + bb*TC*HW + rem;
    float*       py = y + bb*TC*HW + rem;
    float v[TC]; float m = -3.402823466e38f;
    #pragma unroll
    for (int c = 0; c < TC; ++c) { v[c] = px[c*HW]; m = fmaxf(m, v[c]); }
    float s = 0.f;
    #pragma unroll
    for (int c = 0; c < TC; ++c) s += expf(v[c] - m);
    float lse = m + logf(s);
    #pragma unroll
    for (int c = 0; c < TC; ++c) py[c*HW] = v[c] - lse;
}

extern "C" void kernel_launch(void* const* d_in, const int* in_sizes, int n_in,
                              void* d_out, int out_size, void* d_ws, size_t ws_size,
                              hipStream_t stream) {
    const float* x0        = (const float*)d_in[0];  // (B,16,H,W)
    const float* spacings  = (const float*)d_in[1];  // (B,2)
    const float* weight    = (const float*)d_in[2];  // scalar
    const float* inv_theta = (const float*)d_in[3];  // (2,)

    int  B   = in_sizes[1] / 2;
    long chw = (long)in_sizes[0] / B;
    long hw  = chw / TC;
    int  H   = (int)(sqrt((double)hw) + 0.5);
    int  W   = (int)(hw / H);

    float* bufA = (float*)d_ws;     // needs B*16*H*W floats (151 MB for ref shapes)
    float* out  = (float*)d_out;
    float* bufs[2] = { bufA, out };

    dim3 block(256);
    dim3 grid((W + TW - 1)/TW, (H + TH - 1)/TH, B);
    size_t lds = (size_t)(TC*RH*RW + TC*TH*RW) * sizeof(float);  // ~113 KB

    const float* src = x0;
    for (int it = 0; it < 5; ++it) {
        float* dst = bufs[it & 1];   // it0->ws, it1->out, ..., it4->ws
        crf_iter_kernel<<<grid, block, lds, stream>>>(
            src, x0, dst, spacings, inv_theta, weight, H, W);
        src = dst;
    }
    long npix = (long)B * H * W;
    int  nblk = (int)((npix + 255) / 256);
    crf_logsoftmax_kernel<<<nblk, block, 0, stream>>>(src, out, npix, (long)H*W);
}